// DynamicGraphSAGE_81123342287581
// MI455X (gfx1250) — compile-verified
//
#include <hip/hip_runtime.h>

#define N_NODES 10000
#define N_EDGES 640000
#define DFEAT   128

typedef __attribute__((ext_vector_type(2))) float v2f;
typedef __attribute__((ext_vector_type(8))) float v8f;

// ---------------------------------------------------------------------------
// Zero-fill a float buffer.
// ---------------------------------------------------------------------------
__global__ __launch_bounds__(256) void k_zero(float* __restrict__ p, int n) {
  int i = blockIdx.x * 256 + threadIdx.x;
  if (i < n) p[i] = 0.0f;
}

// ---------------------------------------------------------------------------
// Degree histogram: one thread per edge, float atomic add (no-return fadd).
// ---------------------------------------------------------------------------
__global__ __launch_bounds__(256) void k_degree(const long long* __restrict__ dst,
                                                float* __restrict__ deg, int n_edges) {
  int e = blockIdx.x * 256 + threadIdx.x;
  if (e < n_edges) unsafeAtomicAdd(&deg[(int)dst[e]], 1.0f);
}

// ---------------------------------------------------------------------------
// Edge scatter: one wave32 per edge. Each lane moves 4 consecutive floats
// (float4 load, 4x global_atomic_add_f32). Feature table (5.12 MB) is
// L2-resident, so the random gather hits the 192 MB L2, not HBM.
// ---------------------------------------------------------------------------
__global__ __launch_bounds__(256) void k_scatter(const float* __restrict__ feat,
                                                 const long long* __restrict__ src,
                                                 const long long* __restrict__ dst,
                                                 float* __restrict__ agg, int n_edges) {
  int e = blockIdx.x * 8 + (threadIdx.x >> 5);
  if (e >= n_edges) return;
  int lane = threadIdx.x & 31;
  int s = (int)src[e];
  int d = (int)dst[e];
  float4 v = ((const float4*)(feat + (size_t)s * DFEAT))[lane];
  float* ap = agg + (size_t)d * DFEAT + lane * 4;
  unsafeAtomicAdd(ap + 0, v.x);
  unsafeAtomicAdd(ap + 1, v.y);
  unsafeAtomicAdd(ap + 2, v.z);
  unsafeAtomicAdd(ap + 3, v.w);
}

// ---------------------------------------------------------------------------
// Fused SAGE layer: out = ELU( (agg/deg) @ Wl + x @ Wr + b )
// One block = 16 rows of output; wave w owns columns [16w, 16w+16).
// fp32 WMMA 16x16x4 (V_WMMA_F32_16X16X4_F32), 64 WMMAs per wave.
//
// VGPR layouts (ISA 7.12.2, wave32, fp32):
//   A 16x4 : lanes 0-15 -> M=lane, v0/v1 hold K=k0,k0+1;
//            lanes 16-31 -> M=lane-16, v0/v1 hold K=k0+2,k0+3.
//   B 4x16 : lanes 0-15 -> N=lane with K=k0,k0+1; lanes 16-31 -> K=k0+2,k0+3.
//   C/D    : 8 VGPRs, VGPR r -> M=r (lanes 0-15, N=lane) / M=r+8 (lanes 16-31).
// ---------------------------------------------------------------------------
__global__ __launch_bounds__(256) void k_sage_gemm(
    const float* __restrict__ agg,   // [N, D] summed neighbor messages
    const float* __restrict__ deg,   // [N]
    const float* __restrict__ xin,   // [N, D] self features
    const float* __restrict__ Wl,    // [D, D] row-major (k-major)
    const float* __restrict__ Wr,    // [D, D]
    const float* __restrict__ bias,  // [D]
    float* __restrict__ out)         // [N, D]
{
  const int lane = threadIdx.x & 31;
  const int wave = threadIdx.x >> 5;       // 0..7 -> 16-wide column tile
  const int row0 = blockIdx.x * 16;
  const int col0 = wave * 16;
  const int mn   = lane & 15;              // M index for A, N index for B
  const int koff = (lane >> 4) << 1;       // 0 or 2

  const float* arow = agg + (size_t)(row0 + mn) * DFEAT;
  const float* xrow = xin + (size_t)(row0 + mn) * DFEAT;
  const float  invd = 1.0f / fmaxf(deg[row0 + mn], 1.0f);

  v8f acc = {};
#pragma unroll 4
  for (int k0 = 0; k0 < DFEAT; k0 += 4) {
    v2f aA, aX, bL, bR;
    aA[0] = arow[k0 + koff]     * invd;
    aA[1] = arow[k0 + koff + 1] * invd;
    aX[0] = xrow[k0 + koff];
    aX[1] = xrow[k0 + koff + 1];
    bL[0] = Wl[(k0 + koff)     * DFEAT + col0 + mn];
    bL[1] = Wl[(k0 + koff + 1) * DFEAT + col0 + mn];
    bR[0] = Wr[(k0 + koff)     * DFEAT + col0 + mn];
    bR[1] = Wr[(k0 + koff + 1) * DFEAT + col0 + mn];
    // acc += A_agg x Wl_tile
    acc = __builtin_amdgcn_wmma_f32_16x16x4_f32(false, aA, false, bL,
                                                (short)0, acc, false, false);
    // acc += A_x x Wr_tile
    acc = __builtin_amdgcn_wmma_f32_16x16x4_f32(false, aX, false, bR,
                                                (short)0, acc, false, false);
  }

  const float bcol = bias[col0 + mn];
  const int   mb   = (lane >> 4) * 8;
  float* orow = out + (size_t)(row0 + mb) * DFEAT + col0 + mn;
#pragma unroll
  for (int r = 0; r < 8; ++r) {
    float v = acc[r] + bcol;
    v = (v > 0.0f) ? v : expm1f(v);        // ELU(alpha=1)
    orow[(size_t)r * DFEAT] = v;
  }
}

// ---------------------------------------------------------------------------
// Inputs (setup_inputs order):
//  0 x [N*D] f32, 1 edge_index [2*E] i64, 2 W_l0, 3 b0, 4 W_r0,
//  5 W_l1, 6 b1, 7 W_r1.  Output: h [N*D] f32.
// ---------------------------------------------------------------------------
extern "C" void kernel_launch(void* const* d_in, const int* in_sizes, int n_in,
                              void* d_out, int out_size, void* d_ws, size_t ws_size,
                              hipStream_t stream) {
  const float*     x   = (const float*)d_in[0];
  const long long* ei  = (const long long*)d_in[1];
  const float*     Wl0 = (const float*)d_in[2];
  const float*     b0  = (const float*)d_in[3];
  const float*     Wr0 = (const float*)d_in[4];
  const float*     Wl1 = (const float*)d_in[5];
  const float*     b1  = (const float*)d_in[6];
  const float*     Wr1 = (const float*)d_in[7];
  float*           out = (float*)d_out;

  const long long* src = ei;             // edge_index[0, :]
  const long long* dst = ei + N_EDGES;   // edge_index[1, :]

  char*  ws   = (char*)d_ws;
  float* deg  = (float*)ws;                                         // 40 KB
  float* aggb = (float*)(ws + (1 << 16));                           // 5.12 MB
  float* h1   = (float*)(ws + (1 << 16) +
                         (size_t)N_NODES * DFEAT * sizeof(float));  // 5.12 MB

  const int zeroN  = (N_NODES + 255) / 256;
  const int zeroND = (N_NODES * DFEAT + 255) / 256;
  const int edgeT  = (N_EDGES + 255) / 256;
  const int edgeW  = (N_EDGES + 7) / 8;     // 8 waves per 256-thread block
  const int gemmB  = N_NODES / 16;          // 10000 % 16 == 0

  // Degrees (same edge list both layers -> compute once).
  k_zero  <<<zeroN, 256, 0, stream>>>(deg, N_NODES);
  k_degree<<<edgeT, 256, 0, stream>>>(dst, deg, N_EDGES);

  // Layer 1: h1 = ELU(mean_agg(x) @ Wl0 + x @ Wr0 + b0)
  k_zero     <<<zeroND, 256, 0, stream>>>(aggb, N_NODES * DFEAT);
  k_scatter  <<<edgeW, 256, 0, stream>>>(x, src, dst, aggb, N_EDGES);
  k_sage_gemm<<<gemmB, 256, 0, stream>>>(aggb, deg, x, Wl0, Wr0, b0, h1);

  // Layer 2: out = ELU(mean_agg(h1) @ Wl1 + h1 @ Wr1 + b1)
  k_zero     <<<zeroND, 256, 0, stream>>>(aggb, N_NODES * DFEAT);
  k_scatter  <<<edgeW, 256, 0, stream>>>(h1, src, dst, aggb, N_EDGES);
  k_sage_gemm<<<gemmB, 256, 0, stream>>>(aggb, deg, h1, Wl1, Wr1, b1, out);
}